// GCN_Model_74904229642990
// MI455X (gfx1250) — compile-verified
//
#include <hip/hip_runtime.h>

// ---------------------------------------------------------------------------
// 2-layer GCN for gfx1250 (MI455X).
// Bandwidth-bound: edge scatter dominates; GEMMs done with f32 WMMA
// (V_WMMA_F32_16X16X4_F32) for full f32 precision at negligible cost.
// Layer-2 rewritten as (h*norm)@W2 BEFORE aggregation (linearity) to cut
// scatter traffic 4x. Scatter-adds forced to hardware global_atomic_add_f32
// via inline asm (avoids any CAS-loop lowering of atomicrmw fadd).
// ---------------------------------------------------------------------------

typedef __attribute__((ext_vector_type(2))) float v2f;
typedef __attribute__((ext_vector_type(8))) float v8f;

// Native f32 atomic add, no return value (STOREcnt-tracked; drained by the
// implicit S_WAIT_IDLE at S_ENDPGM, so cross-kernel ordering via the stream
// is sufficient).
__device__ __forceinline__ void atomAddF(float* p, float v) {
    asm volatile("global_atomic_add_f32 %0, %1, off" :: "v"(p), "v"(v) : "memory");
}

// ---------------- zero fill -------------------------------------------------
__global__ void k_zero(float* __restrict__ p, long n) {
    long i = (long)blockIdx.x * blockDim.x + threadIdx.x;
    long stride = (long)gridDim.x * blockDim.x;
    for (; i < n; i += stride) p[i] = 0.0f;
}

// ---------------- degree counting ------------------------------------------
__global__ void k_deg(const int* __restrict__ src, const int* __restrict__ dst,
                      int ne, float* __restrict__ degS, float* __restrict__ degD) {
    int e = blockIdx.x * blockDim.x + threadIdx.x;
    if (e < ne) {
        atomAddF(&degS[src[e]], 1.0f);
        atomAddF(&degD[dst[e]], 1.0f);
    }
}

// deg -> rsqrt(max(deg,1)) in place (covers all 4 degree arrays contiguously)
__global__ void k_rsqrt(float* __restrict__ p, int n) {
    int i = blockIdx.x * blockDim.x + threadIdx.x;
    if (i < n) p[i] = rsqrtf(fmaxf(p[i], 1.0f));
}

// ---------------- SpMM layer 1: 128-wide rows, wave32 per edge -------------
// Edge index is wave-uniform by construction so src/dst/norm broadcasts can
// scalarize; each lane moves one float4 (32 lanes * 4 = 128 features).
__global__ void k_spmm128(const float* __restrict__ x, const int* __restrict__ src,
                          const int* __restrict__ dst, const float* __restrict__ normS,
                          float* __restrict__ agg, int ne) {
    int e = blockIdx.x * 8 + (threadIdx.x >> 5);   // 8 waves per 256-thread block
    int lane = threadIdx.x & 31;
    if (e >= ne) return;
    int s = src[e];
    int d = dst[e];
    float ns = normS[s];
    const float4* xp = (const float4*)(x + (long)s * 128);
    float4 v = xp[lane];
    float* ap = agg + (long)d * 128 + lane * 4;
    atomAddF(ap + 0, v.x * ns);
    atomAddF(ap + 1, v.y * ns);
    atomAddF(ap + 2, v.z * ns);
    atomAddF(ap + 3, v.w * ns);
}

// ---------------- SpMM layer 2: 64-wide rows, wave32 per edge --------------
__global__ void k_spmm64(const float* __restrict__ g, const int* __restrict__ src,
                         const int* __restrict__ dst, float* __restrict__ agg, int ne) {
    int e = blockIdx.x * 8 + (threadIdx.x >> 5);
    int lane = threadIdx.x & 31;
    if (e >= ne) return;
    int s = src[e];
    int d = dst[e];
    const float2* gp = (const float2*)(g + (long)s * 64);
    float2 v = gp[lane];
    float* ap = agg + (long)d * 64 + lane * 2;
    atomAddF(ap + 0, v.x);
    atomAddF(ap + 1, v.y);
}

// ---------------- WMMA f32 GEMM: C[M,N] = op(rowscale*A[M,K] @ W[K,N] + b) -
// One wave32 computes one 16x16 output tile, stepping K by 4 with
// V_WMMA_F32_16X16X4_F32 (f32 A/B, f32 accum => full precision).
// Operand layouts per CDNA5 ISA 7.12.2:
//   A 16x4 : lanes 0-15 -> M=lane, VGPR0/1 = K{0,1}; lanes 16-31 = K{2,3}
//   B 4x16 : VGPR0 = row K0 (lanes 0-15 are N) / row K2 (lanes 16-31); VGPR1 = K1/K3
//   C 16x16: VGPR i = M=i (lanes 0-15) / M=8+i (lanes 16-31), N=lane&15
// Grid is exact (M % 16 == 0), no divergence -> EXEC all ones as required.
template <int K, int N, bool RELU, bool BIAS>
__global__ void k_gemm_wmma(const float* __restrict__ A,
                            const float* __restrict__ rowscale,
                            const float* __restrict__ W,
                            const float* __restrict__ bias,
                            float* __restrict__ C) {
    constexpr int tilesN = N / 16;
    int wave = (blockIdx.x * blockDim.x + threadIdx.x) >> 5;
    int lane = threadIdx.x & 31;
    int tm = wave / tilesN;
    int tn = wave % tilesN;

    int half = lane >> 4;      // 0: K pair {0,1}, 1: K pair {2,3}
    int l15  = lane & 15;
    int arow = tm * 16 + l15;  // this lane's A row
    int gcol = tn * 16 + l15;  // this lane's B/C column

    float scale = rowscale[arow];
    const float* ap = A + (long)arow * K + half * 2;
    const float* wp = W + gcol;

    v8f c = {};
    for (int k = 0; k < K; k += 4) {
        v2f a;
        a.x = ap[k]     * scale;     // A[arow][k + half*2 + 0]
        a.y = ap[k + 1] * scale;     // A[arow][k + half*2 + 1]
        int kb = k + half * 2;
        v2f b;
        b.x = wp[(long)kb * N];            // W[kb][gcol]
        b.y = wp[(long)(kb + 1) * N];      // W[kb+1][gcol]
        c = __builtin_amdgcn_wmma_f32_16x16x4_f32(
                /*neg_a=*/false, a, /*neg_b=*/false, b,
                /*c_mod=*/(short)0, c, /*reuse_a=*/false, /*reuse_b=*/false);
    }

    float bval = 0.0f;
    if (BIAS) bval = bias[gcol];
    int rb = half * 8;
    #pragma unroll
    for (int i = 0; i < 8; ++i) {
        float v = c[i] + bval;
        if (RELU) v = fmaxf(v, 0.0f);
        C[(long)(tm * 16 + rb + i) * N + gcol] = v;
    }
}

// ---------------- epilogue: out = agg2 * normD2[row] + b2[col] -------------
__global__ void k_final(const float* __restrict__ agg2, const float* __restrict__ normD2,
                        const float* __restrict__ b2, float* __restrict__ out, int tot) {
    int t = blockIdx.x * blockDim.x + threadIdx.x;
    if (t < tot) {
        int r = t >> 6;      // N_CLS = 64
        int f = t & 63;
        out[t] = agg2[t] * normD2[r] + b2[f];
    }
}

extern "C" void kernel_launch(void* const* d_in, const int* in_sizes, int n_in,
                              void* d_out, int out_size, void* d_ws, size_t ws_size,
                              hipStream_t stream) {
    const float* x    = (const float*)d_in[0];
    const float* W1   = (const float*)d_in[1];
    const float* b1   = (const float*)d_in[2];
    const float* W2   = (const float*)d_in[3];
    const float* b2   = (const float*)d_in[4];
    const int*   src1 = (const int*)d_in[5];
    const int*   dst1 = (const int*)d_in[6];
    const int*   src2 = (const int*)d_in[7];
    const int*   dst2 = (const int*)d_in[8];

    const int IN_F = 128, H_F = 256, N_CLS = 64;
    const int NSRC1 = in_sizes[0] / IN_F;   // 500000
    const int E1    = in_sizes[5];          // 1600000
    const int E2    = in_sizes[7];          // 262144
    const int ND1   = 100000;
    const int ND2   = 16384;

    // workspace layout (floats)
    float* ws    = (float*)d_ws;
    float* degS1 = ws;                           // [NSRC1]
    float* degD1 = degS1 + NSRC1;                // [ND1]
    float* degS2 = degD1 + ND1;                  // [ND1]
    float* degD2 = degS2 + ND1;                  // [ND2]
    float* agg1  = degD2 + ND2;                  // [ND1*128]
    float* agg2  = agg1 + (long)ND1 * IN_F;      // [ND2*64]
    float* h     = agg2 + (long)ND2 * N_CLS;     // [ND1*256]
    float* g     = h + (long)ND1 * H_F;          // [ND1*64]

    // 1) zero degree arrays + both accumulators (contiguous prefix of ws)
    long zcount = (NSRC1 + ND1 + ND1 + ND2) + (long)ND1 * IN_F + (long)ND2 * N_CLS;
    k_zero<<<2048, 256, 0, stream>>>(ws, zcount);

    // 2) degrees + rsqrt norms
    k_deg<<<(E1 + 255) / 256, 256, 0, stream>>>(src1, dst1, E1, degS1, degD1);
    k_deg<<<(E2 + 255) / 256, 256, 0, stream>>>(src2, dst2, E2, degS2, degD2);
    int degtot = NSRC1 + ND1 + ND1 + ND2;
    k_rsqrt<<<(degtot + 255) / 256, 256, 0, stream>>>(ws, degtot);

    // 3) SpMM1: agg1[dst] += x[src] * normS1[src]   (wave per edge, 8/block)
    k_spmm128<<<(E1 + 7) / 8, 256, 0, stream>>>(x, src1, dst1, degS1, agg1, E1);

    // 4) GEMM1 + bias + ReLU: h = relu((agg1 * normD1) @ W1 + b1)
    {
        int waves = (ND1 / 16) * (H_F / 16);   // 6250 * 16 = 100000, 4 waves/block
        k_gemm_wmma<128, 256, true, true>
            <<<waves / 4, 128, 0, stream>>>(agg1, degD1, W1, b1, h);
    }

    // 5) GEMM2 (pre-aggregation, by linearity): g = (h * normS2) @ W2
    {
        int waves = (ND1 / 16) * (N_CLS / 16); // 6250 * 4 = 25000
        k_gemm_wmma<256, 64, false, false>
            <<<waves / 4, 128, 0, stream>>>(h, degS2, W2, nullptr, g);
    }

    // 6) SpMM2: agg2[dst] += g[src]   (wave per edge, 64-wide rows)
    k_spmm64<<<(E2 + 7) / 8, 256, 0, stream>>>(g, src2, dst2, agg2, E2);

    // 7) epilogue: out = agg2 * normD2 + b2
    int tot = ND2 * N_CLS;
    k_final<<<(tot + 255) / 256, 256, 0, stream>>>(agg2, degD2, b2, (float*)d_out, tot);
}